// FusionMamba_46566035423442
// MI455X (gfx1250) — compile-verified
//
#include <hip/hip_runtime.h>
#include <hip/hip_bf16.h>
#include <math.h>

typedef __attribute__((ext_vector_type(2))) float v2f;
typedef __attribute__((ext_vector_type(4))) float v4f;
typedef __attribute__((ext_vector_type(8))) float v8f;

#define D_MODEL 1024
#define D_STATE 16
#define D_CONV  4
#define D_INNER 1024
#define DT_RANK 64
#define BATCH   4
#define SEQLEN  2048
#define NROWS   (BATCH * SEQLEN)   // 8192

// ---------------------------------------------------------------------------
// Generic fp32 WMMA GEMM:  C[M,N] = A[M,K] * W[N,K]^T   (+ epilogue)
// Wave tile: 32(M) x 64(N)  -> acc[2][4] of v8f.
// Block: 128 threads = 4 waves in 2x2 -> block tile 64 x 128.
// Double-buffered K pipeline over 16-wide chunks (K must be a multiple of 32).
// NB  : N not a multiple of the tile -> clamp+mask B rows (branch-free).
// EPI : 0 = store, 1 = bias+softplus, 2 = +residual.
// M must be a multiple of 64.
// ---------------------------------------------------------------------------
template <bool NB, int EPI>
__global__ __launch_bounds__(128)
void gemm_wmma_f32(const float* __restrict__ A, int lda,
                   const float* __restrict__ W, int ldw,
                   float* __restrict__ C, int ldc,
                   int N, int K,
                   const float* __restrict__ bias,    // EPI==1
                   const float* __restrict__ resid,   // EPI==2
                   int ldres)
{
    const int lane = threadIdx.x & 31;
    const int wave = threadIdx.x >> 5;
    const int half = lane >> 4;          // 0: K=k..k+1, 1: K=k+2..k+3
    const int lr   = lane & 15;
    const int wm = wave >> 1, wn = wave & 1;
    const int m_base = blockIdx.y * 64 + wm * 32;
    const int n_base = blockIdx.x * 128 + wn * 64;

    v8f acc[2][4] = {};

    // Hoisted per-lane base pointers (include the half's K offset).
    const float* pa[2];
#pragma unroll
    for (int i = 0; i < 2; ++i)
        pa[i] = A + (size_t)(m_base + 16 * i + lr) * lda + 2 * half;

    const float* pb[4];
    float bmask[4];
#pragma unroll
    for (int j = 0; j < 4; ++j) {
        int row = n_base + 16 * j + lr;
        if (NB) {
            bmask[j] = (row < N) ? 1.0f : 0.0f;
            row = (row < N) ? row : (N - 1);     // clamped -> always in-bounds
        }
        pb[j] = W + (size_t)row * ldw + 2 * half;
    }

    // Load one 16-wide K chunk (4 WMMA steps) into a register buffer.
    auto loadc = [&](int k, v2f (&Ab)[2][4], v2f (&Bb)[4][4]) {
#pragma unroll
        for (int s = 0; s < 4; ++s) {
#pragma unroll
            for (int i = 0; i < 2; ++i)
                Ab[i][s] = *(const v2f*)(pa[i] + k + 4 * s);   // global_load_b64
#pragma unroll
            for (int j = 0; j < 4; ++j) {
                Bb[j][s] = *(const v2f*)(pb[j] + k + 4 * s);
                if (NB) Bb[j][s] = Bb[j][s] * bmask[j];
            }
        }
    };
    auto compute = [&](const v2f (&Ab)[2][4], const v2f (&Bb)[4][4]) {
#pragma unroll
        for (int s = 0; s < 4; ++s)
#pragma unroll
            for (int i = 0; i < 2; ++i)
#pragma unroll
                for (int j = 0; j < 4; ++j)
                    acc[i][j] = __builtin_amdgcn_wmma_f32_16x16x4_f32(
                        false, Ab[i][s], false, Bb[j][s], (short)0,
                        acc[i][j], false, false);
    };

    v2f A0[2][4], B0[4][4], A1[2][4], B1[4][4];
    loadc(0, A0, B0);
    for (int k = 0; k < K; k += 32) {          // K % 32 == 0
        loadc(k + 16, A1, B1);                 // always legal: k <= K-32
        compute(A0, B0);
        if (k + 32 < K) loadc(k + 32, A0, B0);
        compute(A1, B1);
    }

    // Epilogue + store. C/D layout: VGPR e -> M = m_sub + e + 8*half, N = n_sub + lr.
#pragma unroll
    for (int i = 0; i < 2; ++i) {
#pragma unroll
        for (int j = 0; j < 4; ++j) {
            const int n = n_base + 16 * j + lr;
            if (NB && n >= N) continue;
#pragma unroll
            for (int e = 0; e < 8; ++e) {
                const int m = m_base + 16 * i + e + 8 * half;
                float v = acc[i][j][e];
                if (EPI == 1) {                       // bias + softplus
                    v += bias[n];
                    v = (v > 20.0f) ? v : log1pf(__expf(v));
                } else if (EPI == 2) {                // + residual
                    v += resid[(size_t)m * ldres + n];
                }
                C[(size_t)m * ldc + n] = v;
            }
        }
    }
}

// ---------------------------------------------------------------------------
// Depthwise causal conv (width 4) + bias + SiLU.
// Input: xi = first D_INNER columns of xz (row stride 2*D_INNER).
// ---------------------------------------------------------------------------
__global__ __launch_bounds__(256)
void conv_silu_kernel(const float* __restrict__ xz,
                      const float* __restrict__ cw,   // (D_INNER, 1, 4)
                      const float* __restrict__ cb,
                      float* __restrict__ xi_out)
{
    const int idx = blockIdx.x * 256 + threadIdx.x;       // 0 .. NROWS*D_INNER-1
    const int d   = idx & (D_INNER - 1);
    const int row = idx >> 10;
    const int l   = row & (SEQLEN - 1);
    const int b   = row >> 11;

    float s = cb[d];
#pragma unroll
    for (int j = 0; j < D_CONV; ++j) {
        const int lsrc = l - (D_CONV - 1) + j;
        if (lsrc >= 0)
            s += cw[d * D_CONV + j] *
                 xz[((size_t)(b * SEQLEN + lsrc)) * (2 * D_INNER) + d];
    }
    s = s / (1.0f + __expf(-s));   // silu
    xi_out[idx] = s;
}

// ---------------------------------------------------------------------------
// h0[b, i] = dot(space_cond[b, :], hidden_proj_w[i, :])   (one wave / output)
// ---------------------------------------------------------------------------
__global__ __launch_bounds__(256)
void h0_kernel(const float* __restrict__ sc,
               const float* __restrict__ w,
               float* __restrict__ h)
{
    const int wg   = (blockIdx.x * 256 + threadIdx.x) >> 5; // 0..65535
    const int lane = threadIdx.x & 31;
    const int b    = wg >> 14;
    const int i    = wg & 16383;

    const float* wr = w + (size_t)i * D_MODEL;
    const float* s  = sc + b * D_MODEL;
    float sum = 0.0f;
    for (int k = lane; k < D_MODEL; k += 32) sum += wr[k] * s[k];
#pragma unroll
    for (int off = 16; off > 0; off >>= 1) sum += __shfl_down(sum, off, 32);
    if (lane == 0) h[wg] = sum;
}

// ---------------------------------------------------------------------------
// Selective scan: one thread per (b, d); 16 states in registers.
// Fuses +u*D and the SiLU gate; writes y_gated into xz's first half.
// B/C rows are read as 4x b128 vectors each (wave-uniform, L2-hot).
// ---------------------------------------------------------------------------
__global__ __launch_bounds__(256)
void scan_kernel(const float* __restrict__ xi,
                 const float* __restrict__ delta,
                 const float* __restrict__ xdbl,   // (rows, 96): [64:80]=B, [80:96]=C
                 const float* __restrict__ A_log,
                 const float* __restrict__ Dp,
                 float* __restrict__ xz,           // read res (cols 1024..), write y (cols 0..)
                 float* __restrict__ h)            // in: h0, out: h_last
{
    const int gid = blockIdx.x * 256 + threadIdx.x;  // 0..4095
    const int b = gid >> 10;
    const int d = gid & (D_INNER - 1);

    float Av[D_STATE], hh[D_STATE];
#pragma unroll
    for (int n = 0; n < D_STATE; ++n) {
        Av[n] = -__expf(A_log[d * D_STATE + n]);
        hh[n] = h[(size_t)gid * D_STATE + n];
    }
    const float Dd = Dp[d];
    const size_t rowbase = (size_t)b * SEQLEN;

    for (int l = 0; l < SEQLEN; ++l) {
        const size_t row = rowbase + l;
        const float dt = delta[row * D_INNER + d];
        const float u  = xi[row * D_INNER + d];
        const float du = dt * u;
        const float* bc = xdbl + row * (DT_RANK + 2 * D_STATE);
        v4f Bv[4], Cv[4];
#pragma unroll
        for (int q = 0; q < 4; ++q) {
            Bv[q] = *(const v4f*)(bc + DT_RANK + 4 * q);              // b128
            Cv[q] = *(const v4f*)(bc + DT_RANK + D_STATE + 4 * q);    // b128
        }
        float y = 0.0f;
#pragma unroll
        for (int n = 0; n < D_STATE; ++n) {
            const float dA = __expf(dt * Av[n]);
            hh[n] = dA * hh[n] + du * Bv[n >> 2][n & 3];
            y += hh[n] * Cv[n >> 2][n & 3];
        }
        y += u * Dd;
        const float r = xz[row * (2 * D_INNER) + D_INNER + d];
        y *= r / (1.0f + __expf(-r));                 // * silu(res)
        xz[row * (2 * D_INNER) + d] = y;              // y_gated (reuse xi slot)
    }
#pragma unroll
    for (int n = 0; n < D_STATE; ++n) h[(size_t)gid * D_STATE + n] = hh[n];
}

// ---------------------------------------------------------------------------
// LayerNorm over D_MODEL per row. One block (256 thr) per row.
// ---------------------------------------------------------------------------
__global__ __launch_bounds__(256)
void layernorm_kernel(const float* __restrict__ x,
                      const float* __restrict__ gamma,
                      const float* __restrict__ beta,
                      float* __restrict__ out)
{
    __shared__ float red[256];
    const int row = blockIdx.x;
    const int t   = threadIdx.x;
    const float* xr = x + (size_t)row * D_MODEL;

    float s = 0.0f;
    for (int k = t; k < D_MODEL; k += 256) s += xr[k];
    red[t] = s; __syncthreads();
    for (int off = 128; off > 0; off >>= 1) {
        if (t < off) red[t] += red[t + off];
        __syncthreads();
    }
    const float mu = red[0] * (1.0f / D_MODEL);
    __syncthreads();

    float v = 0.0f;
    for (int k = t; k < D_MODEL; k += 256) { float dd = xr[k] - mu; v += dd * dd; }
    red[t] = v; __syncthreads();
    for (int off = 128; off > 0; off >>= 1) {
        if (t < off) red[t] += red[t + off];
        __syncthreads();
    }
    const float rstd = rsqrtf(red[0] * (1.0f / D_MODEL) + 1e-5f);

    for (int k = t; k < D_MODEL; k += 256)
        out[(size_t)row * D_MODEL + k] = (xr[k] - mu) * rstd * gamma[k] + beta[k];
}

// ---------------------------------------------------------------------------
// hist[b, j] = dot(h_last[b, :16384], history_w[j, :16384]); block per output.
// ---------------------------------------------------------------------------
__global__ __launch_bounds__(256)
void hist_kernel(const float* __restrict__ h,
                 const float* __restrict__ w,
                 float* __restrict__ out)
{
    __shared__ float red[256];
    const int o = blockIdx.x;            // b*1024 + j
    const int b = o >> 10;
    const int j = o & 1023;
    const int t = threadIdx.x;

    const float* hr = h + (size_t)b * (D_INNER * D_STATE);
    const float* wr = w + (size_t)j * (D_INNER * D_STATE);
    float s = 0.0f;
    for (int k = t; k < D_INNER * D_STATE; k += 256) s += hr[k] * wr[k];
    red[t] = s; __syncthreads();
    for (int off = 128; off > 0; off >>= 1) {
        if (t < off) red[t] += red[t + off];
        __syncthreads();
    }
    if (t == 0) out[o] = red[0];
}

// ---------------------------------------------------------------------------
extern "C" void kernel_launch(void* const* d_in, const int* in_sizes, int n_in,
                              void* d_out, int out_size, void* d_ws, size_t ws_size,
                              hipStream_t stream) {
    const float* x       = (const float*)d_in[0];   // (4,2048,1024)
    const float* sc      = (const float*)d_in[1];   // (4,1,1024)
    const float* in_w    = (const float*)d_in[2];   // (2048,1024)
    const float* conv_w  = (const float*)d_in[3];   // (1024,1,4)
    const float* conv_b  = (const float*)d_in[4];
    const float* xproj_w = (const float*)d_in[5];   // (96,1024)
    const float* dt_w    = (const float*)d_in[6];   // (1024,64)
    const float* dt_b    = (const float*)d_in[7];
    const float* A_log   = (const float*)d_in[8];   // (1024,16)
    const float* Dp      = (const float*)d_in[9];
    const float* hid_w   = (const float*)d_in[10];  // (16384,1024)
    const float* hist_w  = (const float*)d_in[11];  // (1024,16384)
    const float* out_w   = (const float*)d_in[12];  // (1024,1024)
    const float* gamma   = (const float*)d_in[13];
    const float* beta    = (const float*)d_in[14];
    float* out = (float*)d_out;

    // workspace layout (floats)
    float* ws      = (float*)d_ws;
    float* xz      = ws;                               // 8192 x 2048; cols 0..1023 reused for y_gated
    float* xi_conv = xz      + (size_t)NROWS * 2048;   // 8192 x 1024; reused for out_pre
    float* x_dbl   = xi_conv + (size_t)NROWS * 1024;   // 8192 x 96
    float* delta   = x_dbl   + (size_t)NROWS * 96;     // 8192 x 1024
    float* h       = delta   + (size_t)NROWS * 1024;   // 4 x 16384
    float* out_pre = xi_conv;

    const dim3 blk128(128), blk256(256);

    // 1) xz = x @ in_proj_w^T                         (M=8192, N=2048, K=1024)
    gemm_wmma_f32<false, 0><<<dim3(2048 / 128, NROWS / 64), blk128, 0, stream>>>(
        x, D_MODEL, in_w, D_MODEL, xz, 2 * D_INNER,
        2 * D_INNER, D_MODEL, nullptr, nullptr, 0);

    // 2) depthwise conv + bias + silu
    conv_silu_kernel<<<(NROWS * D_INNER) / 256, blk256, 0, stream>>>(
        xz, conv_w, conv_b, xi_conv);

    // 3) x_dbl = xi @ x_proj_w^T                      (N=96, K=1024)
    gemm_wmma_f32<true, 0><<<dim3(1, NROWS / 64), blk128, 0, stream>>>(
        xi_conv, D_INNER, xproj_w, D_INNER, x_dbl, DT_RANK + 2 * D_STATE,
        DT_RANK + 2 * D_STATE, D_INNER, nullptr, nullptr, 0);

    // 4) delta = softplus(x_dbl[:, :64] @ dt_proj_w^T + b)   (N=1024, K=64)
    gemm_wmma_f32<false, 1><<<dim3(1024 / 128, NROWS / 64), blk128, 0, stream>>>(
        x_dbl, DT_RANK + 2 * D_STATE, dt_w, DT_RANK, delta, D_INNER,
        D_INNER, DT_RANK, dt_b, nullptr, 0);

    // 5) h0 = sc @ hidden_proj_w^T   (one wave per output, 65536 waves)
    h0_kernel<<<(BATCH * D_INNER * D_STATE * 32) / 256, blk256, 0, stream>>>(
        sc, hid_w, h);

    // 6) selective scan (+ u*D, * silu(res)) -> y_gated into xz cols 0..1023
    scan_kernel<<<(BATCH * D_INNER) / 256, blk256, 0, stream>>>(
        xi_conv, delta, x_dbl, A_log, Dp, xz, h);

    // 7) out_pre = y_gated @ out_proj_w^T + x        (N=1024, K=1024)
    gemm_wmma_f32<false, 2><<<dim3(1024 / 128, NROWS / 64), blk128, 0, stream>>>(
        xz, 2 * D_INNER, out_w, D_INNER, out_pre, D_MODEL,
        D_MODEL, D_INNER, nullptr, x, D_MODEL);

    // 8) layernorm -> d_out[0 : 8192*1024]
    layernorm_kernel<<<NROWS, blk256, 0, stream>>>(out_pre, gamma, beta, out);

    // 9) hist = h_last @ history_w^T -> d_out[8192*1024 : +4096]
    hist_kernel<<<BATCH * D_MODEL, blk256, 0, stream>>>(
        h, hist_w, out + (size_t)NROWS * D_MODEL);
}